// HoGLayer_66374424592931
// MI455X (gfx1250) — compile-verified
//
#include <hip/hip_runtime.h>
#include <hip/hip_bf16.h>
#include <math.h>

#define IMG_H 4096
#define IMG_W 4096
#define CELLS 512            // 4096 / 8 cells per side
#define NB 9                 // histogram bins
#define BLOCKS 511           // (CELLS-1) blocks per side
#define HWSZ ((size_t)IMG_H * (size_t)IMG_W)

// ---- CDNA5 async global->LDS path (probe-guarded; falls back to plain loads) ----
#if defined(__has_builtin)
#if __has_builtin(__builtin_amdgcn_global_load_async_to_lds_b32) && \
    __has_builtin(__builtin_amdgcn_s_wait_asynccnt)
#define HAVE_ASYNC_LDS 1
#endif
#endif

// HIP prints address_space(1) as '__device__': builtin wants (as1 int*, as3 int*, imm, imm)
typedef __attribute__((address_space(1))) int GI;   // global int*
typedef __attribute__((address_space(3))) int LI;   // LDS int*

// ---------------------------------------------------------------------------
// Kernel 1: per-cell 9-bin histogram.  One thread = one cell (512x512 cells).
// Each cell needs s() at 4 neighbor pixels; s = sum of 3 channels => 12 loads.
// The 12 gather-loads are issued as async global->LDS DMA (ASYNCcnt-tracked),
// then the thread waits once and reduces from its private LDS slot.
// ---------------------------------------------------------------------------
__global__ __launch_bounds__(256)
void hog_cell_hist(const float* __restrict__ x, float* __restrict__ hist)
{
    const int g  = blockIdx.x * 256 + threadIdx.x;   // cell id 0..262143
    const int cx = g & (CELLS - 1);
    const int cy = g >> 9;
    const int r  = cy * 8 + 7;                       // sampled pixel row
    const int c  = cx * 8 + 7;                       // sampled pixel col
    const bool vmask = (r + 1) < IMG_H;              // zero-padded boundary
    const bool hmask = (c + 1) < IMG_W;
    const int rp = vmask ? (r + 1) : r;              // clamped safe addresses
    const int cp = hmask ? (c + 1) : c;

    float vU, vD, vL, vR;                            // s at (r-1,c),(r+1,c),(r,c-1),(r,c+1)

#ifdef HAVE_ASYNC_LDS
    __shared__ float stage[256 * 12];                // 12 KB staging, per-thread slots
    float* st = &stage[threadIdx.x * 12];
    #pragma unroll
    for (int ch = 0; ch < 3; ++ch) {
        const float* base = x + (size_t)ch * HWSZ;
        __builtin_amdgcn_global_load_async_to_lds_b32(
            (GI*)(base + (size_t)(r - 1) * IMG_W + c),  (LI*)&st[ch * 4 + 0], 0, 0);
        __builtin_amdgcn_global_load_async_to_lds_b32(
            (GI*)(base + (size_t)rp * IMG_W + c),       (LI*)&st[ch * 4 + 1], 0, 0);
        __builtin_amdgcn_global_load_async_to_lds_b32(
            (GI*)(base + (size_t)r * IMG_W + (c - 1)),  (LI*)&st[ch * 4 + 2], 0, 0);
        __builtin_amdgcn_global_load_async_to_lds_b32(
            (GI*)(base + (size_t)r * IMG_W + cp),       (LI*)&st[ch * 4 + 3], 0, 0);
    }
    __builtin_amdgcn_s_wait_asynccnt(0);
    asm volatile("" ::: "memory");
    vU = st[0] + st[4] + st[8];
    vD = st[1] + st[5] + st[9];
    vL = st[2] + st[6] + st[10];
    vR = st[3] + st[7] + st[11];
#else
    vU = vD = vL = vR = 0.0f;
    #pragma unroll
    for (int ch = 0; ch < 3; ++ch) {
        const float* base = x + (size_t)ch * HWSZ;
        vU += base[(size_t)(r - 1) * IMG_W + c];
        vD += base[(size_t)rp * IMG_W + c];
        vL += base[(size_t)r * IMG_W + (c - 1)];
        vR += base[(size_t)r * IMG_W + cp];
    }
#endif
    if (!vmask) vD = 0.0f;                           // zero padding beyond image
    if (!hmask) vR = 0.0f;

    const float gv  = vD - vU;
    const float gh  = vR - vL;
    const float mag = sqrtf(gv * gv + gh * gh + 1e-6f);
    const float ang = fabsf(atanf(gh / (gv + 1e-9f)) * 57.29577951308232f);

    const float jf  = floorf(ang / 20.0f - 0.5f);    // bin index (may be -1)
    const float vj  = mag * ((20.0f * (jf + 1.5f) - ang) / 20.0f);
    const float vj1 = mag - vj;
    const int ji  = (int)jf;                         // in [-1, 4]
    const int i0  = ((ji % NB) + NB) % NB;
    const int i1  = (((ji + 1) % NB) + NB) % NB;

    float* out = hist + (size_t)g * NB;
    #pragma unroll
    for (int b = 0; b < NB; ++b) {
        float v = 0.0f;
        if (b == i0) v += vj;
        if (b == i1) v += vj1;
        out[b] = v;
    }
}

// ---------------------------------------------------------------------------
// Kernel 2: 2x2-cell block assembly + L2 normalization.
// 288 threads = 8 blocks x 36 elements; deterministic serial reduction in LDS.
// ---------------------------------------------------------------------------
__global__ __launch_bounds__(288)
void hog_block_norm(const float* __restrict__ hist, float* __restrict__ feat, int nblk)
{
    __shared__ float sq[288];
    __shared__ float ssum[8];

    const int t  = threadIdx.x;
    const int bl = t / 36;                 // local block 0..7
    const int e  = t - bl * 36;            // element within block 0..35
    const int B  = blockIdx.x * 8 + bl;    // global block id
    const bool active = (B < nblk);

    float val = 0.0f;
    if (active) {
        const int i   = B / BLOCKS;
        const int j   = B - i * BLOCKS;
        const int q   = e / 9;             // quadrant: (i,j),(i,j+1),(i+1,j),(i+1,j+1)
        const int bin = e - q * 9;
        const int ci  = i + (q >> 1);
        const int cj  = j + (q & 1);
        val = hist[((size_t)ci * CELLS + cj) * NB + bin];
    }
    sq[t] = val * val;
    __syncthreads();
    if (e == 0) {                          // fixed-order sum => deterministic
        float s = 0.0f;
        #pragma unroll
        for (int k = 0; k < 36; ++k) s += sq[bl * 36 + k];
        ssum[bl] = s;
    }
    __syncthreads();
    if (active) {
        const float norm = sqrtf(ssum[bl]);
        feat[(size_t)B * 36 + e] = val / (norm + 1e-9f);
    }
}

// ---------------------------------------------------------------------------
extern "C" void kernel_launch(void* const* d_in, const int* in_sizes, int n_in,
                              void* d_out, int out_size, void* d_ws, size_t ws_size,
                              hipStream_t stream)
{
    (void)in_sizes; (void)n_in; (void)out_size; (void)ws_size;
    const float* x  = (const float*)d_in[0];
    float* hist     = (float*)d_ws;        // 512*512*9 floats = 9.4 MB scratch
    float* feat     = (float*)d_out;       // 511*511*36 floats

    hog_cell_hist<<<(CELLS * CELLS) / 256, 256, 0, stream>>>(x, hist);

    const int nblk = BLOCKS * BLOCKS;      // 261121
    hog_block_norm<<<(nblk + 7) / 8, 288, 0, stream>>>(hist, feat, nblk);
}